// Forward_7670811590672
// MI455X (gfx1250) — compile-verified
//
#include <hip/hip_runtime.h>

// y[b] = x[pairs[b,0]]^T * Asym[scheme[b]] * x[pairs[b,1]],  Asym = (A+A^T)/2
// Strategy: bin items by scheme, then per-scheme tiled GEMM with
// V_WMMA_F32_16X16X4_F32 (exact f32 path on CDNA5).

typedef float v2f __attribute__((ext_vector_type(2)));
typedef float v8f __attribute__((ext_vector_type(8)));

#define DIM 64
#define AS_STRIDE 80   // 64+16: half-wave row pairs land in disjoint bank groups
#define X_STRIDE  65   // odd stride: column reads hit 16 distinct banks

// ---------------- binning kernels ----------------

__global__ void k_init(int* ws) {
    if (threadIdx.x < 256) ws[threadIdx.x] = 0;
}

__global__ void k_hist(const int* __restrict__ sidx, int* __restrict__ cnt, int B) {
    int b = blockIdx.x * blockDim.x + threadIdx.x;
    if (b < B) atomicAdd(&cnt[sidx[b]], 1);
}

__global__ void k_scan(int* ws, int NS) {
    // ws: cnt[0..63], cursor at +64, offs at +128
    if (threadIdx.x == 0) {
        int run = 0;
        for (int s = 0; s < NS; ++s) {
            ws[128 + s] = run;   // offs
            ws[64 + s]  = run;   // cursor
            run += ws[s];
        }
    }
}

__global__ void k_scatter(const int* __restrict__ sidx, int* __restrict__ cursor,
                          int* __restrict__ perm, int B) {
    int b = blockIdx.x * blockDim.x + threadIdx.x;
    if (b < B) {
        int pos = atomicAdd(&cursor[sidx[b]], 1);
        perm[pos] = b;
    }
}

// ---------------- main compute ----------------
// block = 128 threads (4 waves). grid = (NS, TILE_SLOTS).
// Each block: load Asym[s] to LDS once; loop over 16-item tiles of its scheme.
// Wave w owns output columns [16w,16w+16): T = Xv(16x64) @ Asym(64x64) via
// 16 chained v_wmma_f32_16x16x4_f32, then y[m] = sum_e T[m][e]*Xu[m][e].

__global__ __launch_bounds__(128)
void k_main(const int* __restrict__ pairs, const float* __restrict__ x,
            const float* __restrict__ A, const int* __restrict__ cnt,
            const int* __restrict__ offs, const int* __restrict__ perm,
            float* __restrict__ out) {
    __shared__ float As[DIM * AS_STRIDE];
    __shared__ float Xv[16 * X_STRIDE];
    __shared__ float Xu[16 * X_STRIDE];
    __shared__ float part[4][16];
    __shared__ int   tI[16];

    const int s    = blockIdx.x;
    const int n    = cnt[s];
    const int base = offs[s];
    const int tid  = threadIdx.x;
    const int wave = tid >> 5;
    const int lane = tid & 31;
    const int half = lane >> 4;      // 0: lanes 0-15, 1: lanes 16-31
    const int l16  = lane & 15;

    // symmetrize A[s] into LDS:  As[d][e] = 0.5*(A[d][e] + A[e][d])
    const float* Ab = A + (size_t)s * DIM * DIM;
    for (int i = tid; i < DIM * DIM; i += 128) {
        int d = i >> 6, e = i & 63;
        As[d * AS_STRIDE + e] = 0.5f * (Ab[i] + Ab[e * DIM + d]);
    }
    __syncthreads();

    for (int t0 = blockIdx.y * 16; t0 < n; t0 += gridDim.y * 16) {
        // ---- gather 16 xv rows and 16 xu rows into LDS ----
        // tid -> row m = tid>>3, 8-float chunk c = (tid&7)*8
        {
            int m = tid >> 3;
            int c = (tid & 7) * 8;
            int bidx = (t0 + m < n) ? perm[base + t0 + m] : -1;
            if (tid < 16) tI[tid] = (t0 + tid < n) ? perm[base + t0 + tid] : -1;
            float* pv = &Xv[m * X_STRIDE + c];
            float* pu = &Xu[m * X_STRIDE + c];
            if (bidx >= 0) {
                int vi = pairs[2 * bidx];
                int ui = pairs[2 * bidx + 1];
                const float* xv = x + (size_t)vi * DIM + c;
                const float* xu = x + (size_t)ui * DIM + c;
                float4 a0 = *(const float4*)(xv);
                float4 a1 = *(const float4*)(xv + 4);
                float4 b0 = *(const float4*)(xu);
                float4 b1 = *(const float4*)(xu + 4);
                pv[0]=a0.x; pv[1]=a0.y; pv[2]=a0.z; pv[3]=a0.w;
                pv[4]=a1.x; pv[5]=a1.y; pv[6]=a1.z; pv[7]=a1.w;
                pu[0]=b0.x; pu[1]=b0.y; pu[2]=b0.z; pu[3]=b0.w;
                pu[4]=b1.x; pu[5]=b1.y; pu[6]=b1.z; pu[7]=b1.w;
            } else {
                #pragma unroll
                for (int j = 0; j < 8; ++j) { pv[j] = 0.0f; pu[j] = 0.0f; }
            }
        }
        __syncthreads();

        // ---- WMMA: T-tile for columns [16*wave, 16*wave+16) ----
        v8f acc = {0.f, 0.f, 0.f, 0.f, 0.f, 0.f, 0.f, 0.f};
        const int ncol = wave * 16 + l16;
        #pragma unroll
        for (int k = 0; k < DIM; k += 4) {
            const int ka = k + 2 * half;
            v2f a, b;
            // A operand (16x4): lane -> M=l16, K = ka, ka+1
            a.x = Xv[l16 * X_STRIDE + ka];
            a.y = Xv[l16 * X_STRIDE + ka + 1];
            // B operand (4x16): VGPR0 -> row ka, VGPR1 -> row ka+1, col = ncol
            b.x = As[ka * AS_STRIDE + ncol];
            b.y = As[(ka + 1) * AS_STRIDE + ncol];
            acc = __builtin_amdgcn_wmma_f32_16x16x4_f32(
                false, a, false, b, (short)0, acc, false, false);
        }

        // ---- y[m] = sum_e T[m][e] * Xu[m][e] ----
        // acc[r] holds T[r + 8*half][ncol]
        float p[8];
        #pragma unroll
        for (int r = 0; r < 8; ++r) {
            int mm = r + 8 * half;
            p[r] = acc[r] * Xu[mm * X_STRIDE + ncol];
        }
        #pragma unroll
        for (int off = 1; off < 16; off <<= 1) {
            #pragma unroll
            for (int r = 0; r < 8; ++r)
                p[r] += __shfl_xor(p[r], off, 16);
        }
        if (l16 == 0) {
            #pragma unroll
            for (int r = 0; r < 8; ++r)
                part[wave][r + 8 * half] = p[r];
        }
        __syncthreads();

        if (tid < 16) {
            int b = tI[tid];
            if (b >= 0)
                out[b] = part[0][tid] + part[1][tid] + part[2][tid] + part[3][tid];
        }
        __syncthreads();  // protect LDS tiles before next iteration
    }
}

// ---------------- launcher ----------------

extern "C" void kernel_launch(void* const* d_in, const int* in_sizes, int n_in,
                              void* d_out, int out_size, void* d_ws, size_t ws_size,
                              hipStream_t stream) {
    const int*   pairs = (const int*)d_in[0];     // (B,2) int32
    const int*   sidx  = (const int*)d_in[1];     // (B,)  int32
    const float* x     = (const float*)d_in[2];   // (NT,64) f32
    const float* A     = (const float*)d_in[3];   // (NS,64,64) f32
    float*       out   = (float*)d_out;

    const int B  = in_sizes[1];
    const int NS = in_sizes[3] / (DIM * DIM);

    int* ws     = (int*)d_ws;
    int* cnt    = ws;          // 64
    int* cursor = ws + 64;     // 64
    int* offs   = ws + 128;    // 64
    int* perm   = ws + 256;    // B

    k_init<<<1, 256, 0, stream>>>(ws);
    k_hist<<<(B + 255) / 256, 256, 0, stream>>>(sidx, cnt, B);
    k_scan<<<1, 32, 0, stream>>>(ws, NS);
    k_scatter<<<(B + 255) / 256, 256, 0, stream>>>(sidx, cursor, perm, B);

    // ~1024 items/scheme expected -> ~64 tiles; 96 slots + stride loop for skew
    dim3 grid(NS, 96);
    k_main<<<grid, 128, 0, stream>>>(pairs, x, A, cnt, offs, perm, out);
}